// RankingHingeLoss_40355512713936
// MI455X (gfx1250) — compile-verified
//
#include <hip/hip_runtime.h>

typedef __attribute__((ext_vector_type(2))) float v2f;
typedef __attribute__((ext_vector_type(8))) float v8f;

#define RH_MARGIN 1.0f
#define RH_NCOL 512
#define RH_NROW 256

// One block per batch row. 256 threads = 8 wave32 waves.
// Pair matrix per row = 512x512 = 32x32 tiles of 16x16 pairs.
// Mask folded into the matmul (binary masks commute with relu):
//   r_i*nr_j*(m - s_i + s_j) = [r_i*(m-s_i)]*nr_j + r_i*[nr_j*s_j]
// A(16x4) = [r_i*(m-s_i), r_i, 0, 0], B(4x16) = [nr_j; nr_j*s_j; 0; 0], C = 0
// => one v_wmma_f32_16x16x4_f32 yields the fully masked hinge argument;
// per element: relu (fmaxf) + serial accumulate (keeps all 8 D elements
// live and prevents the SLP accumulator collapse seen in round 3).
__global__ void __launch_bounds__(256)
rank_hinge_row_kernel(const float* __restrict__ scores,
                      const int* __restrict__ relev,
                      float* __restrict__ partial_sum,
                      unsigned int* __restrict__ partial_cnt)
{
    __shared__ float s_ap[2 * RH_NCOL];   // (r_i*(m-s_i), r_i) pairs
    __shared__ float s_bp[2 * RH_NCOL];   // (nr_j, nr_j*s_j) pairs
    __shared__ float s_zero[2 * RH_NCOL]; // zeros for lanes holding K=2,3
    __shared__ float s_red[256];
    __shared__ unsigned int s_cnt[256];

    const int row = blockIdx.x;
    const int tid = threadIdx.x;
    const float* srow = scores + row * RH_NCOL;
    const int* rrow = relev + row * RH_NCOL;

    unsigned int my_rel = 0;
    for (int c = tid; c < RH_NCOL; c += 256) {
        float s = srow[c];
        int r = rrow[c];
        float rf = (r > 0) ? 1.0f : 0.0f;
        float nrf = 1.0f - rf;
        s_ap[2 * c]     = rf * (RH_MARGIN - s);
        s_ap[2 * c + 1] = rf;
        s_bp[2 * c]     = nrf;
        s_bp[2 * c + 1] = nrf * s;
        s_zero[2 * c]     = 0.0f;
        s_zero[2 * c + 1] = 0.0f;
        my_rel += (r > 0) ? 1u : 0u;
    }
    s_cnt[tid] = my_rel;
    __syncthreads();

    const int lane = tid & 31;          // wave32 on gfx1250
    const int wave = tid >> 5;          // 8 waves
    const int half = (lane >> 4) & 1;   // 0: lanes 0-15 (K=0,1), 1: lanes 16-31 (K=2,3)
    const int lcol = lane & 15;

    // Lanes 16-31 supply the zero K=2,3 columns/rows of A and B:
    // point them at the zeroed LDS region (branch-free address select).
    const float* abase = (half == 0) ? s_ap : s_zero;
    const float* bbase = (half == 0) ? s_bp : s_zero;

    float acc = 0.0f;

    // 32 ti-strips over 8 waves; 32 tj tiles each.
    for (int ti = wave; ti < 32; ti += 8) {
        v2f a = *(const v2f*)(abase + 2 * (ti * 16 + lcol));  // ds_load_b64

        for (int tj = 0; tj < 32; ++tj) {
            v2f b = *(const v2f*)(bbase + 2 * (tj * 16 + lcol));  // ds_load_b64

            v8f c;
            #pragma unroll
            for (int v = 0; v < 8; ++v) c[v] = 0.0f;

            // D[i][j] = r_i*nr_j*(margin - s_i + s_j), 16x16 pairs per wmma
            v8f d = __builtin_amdgcn_wmma_f32_16x16x4_f32(
                /*neg_a=*/false, a, /*neg_b=*/false, b,
                /*c_mod=*/(short)0, c, /*reuse_a=*/false, /*reuse_b=*/false);

            float tsum = 0.0f;
            #pragma unroll
            for (int v = 0; v < 8; ++v)
                tsum += fmaxf(d[v], 0.0f);
            acc += tsum;
        }
    }

    // Deterministic block tree reduction of loss sum and relevant count.
    s_red[tid] = acc;
    __syncthreads();
    for (int off = 128; off > 0; off >>= 1) {
        if (tid < off) {
            s_red[tid] += s_red[tid + off];
            s_cnt[tid] += s_cnt[tid + off];
        }
        __syncthreads();
    }
    if (tid == 0) {
        partial_sum[row] = s_red[0];
        unsigned int nrel = s_cnt[0];
        partial_cnt[row] = nrel * (unsigned int)(RH_NCOL - nrel);
    }
}

// Single block: deterministically reduce 256 per-row partials, divide.
__global__ void __launch_bounds__(256)
rank_hinge_final_kernel(const float* __restrict__ partial_sum,
                        const unsigned int* __restrict__ partial_cnt,
                        float* __restrict__ out)
{
    __shared__ float s_red[256];
    __shared__ unsigned int s_cnt[256];
    const int tid = threadIdx.x;
    s_red[tid] = partial_sum[tid];
    s_cnt[tid] = partial_cnt[tid];
    __syncthreads();
    for (int off = 128; off > 0; off >>= 1) {
        if (tid < off) {
            s_red[tid] += s_red[tid + off];
            s_cnt[tid] += s_cnt[tid + off];
        }
        __syncthreads();
    }
    if (tid == 0) {
        unsigned int n = s_cnt[0];
        out[0] = (n > 0) ? (s_red[0] / (float)n) : 0.0f;
    }
}

extern "C" void kernel_launch(void* const* d_in, const int* in_sizes, int n_in,
                              void* d_out, int out_size, void* d_ws, size_t ws_size,
                              hipStream_t stream) {
    (void)in_sizes; (void)n_in; (void)out_size; (void)ws_size;
    const float* scores = (const float*)d_in[0];
    const int* relev = (const int*)d_in[1];

    float* psum = (float*)d_ws;                                  // 256 floats
    unsigned int* pcnt = (unsigned int*)((char*)d_ws + RH_NROW * sizeof(float)); // 256 uints

    rank_hinge_row_kernel<<<RH_NROW, 256, 0, stream>>>(scores, relev, psum, pcnt);
    rank_hinge_final_kernel<<<1, 256, 0, stream>>>(psum, pcnt, (float*)d_out);
}